// GraphTransformer_21440476741737
// MI455X (gfx1250) — compile-verified
//
#include <hip/hip_runtime.h>
#include <stdint.h>

typedef __attribute__((ext_vector_type(2))) float v2f;
typedef __attribute__((ext_vector_type(8))) float v8f;

#define B_   16
#define N_   1024
#define DIN_ 128
#define DH_  256
#define NP1  1025

// ---------------------------------------------------------------------------
// Kernel 1: bias_mlp on the 6 codebook rows via V_WMMA_F32_16X16X4_F32.
// One wave (32 threads). A = codebook padded to 16x128 (rows 6..15 zero),
// B = W1 (128x256) in 16 column tiles, C/D = f32 16x16 accumulator.
// A-layout (ISA 7.12.2, 32-bit A 16x4): lanes 0-15 -> M=lane, VGPR0=K0,VGPR1=K1;
// lanes 16-31 -> M=lane-16, VGPR0=K2, VGPR1=K3. B mirrors with N in lanes.
// ---------------------------------------------------------------------------
__global__ __launch_bounds__(32)
void mlp6_wmma(const float* __restrict__ cb, const float* __restrict__ W1,
               const float* __restrict__ b1, const float* __restrict__ W2,
               const float* __restrict__ b2, float* __restrict__ s_out) {
  const int lane = threadIdx.x;
  const int half = lane >> 4;   // 0: K pair {0,1} / rows 0-7 of D ; 1: K pair {2,3} / rows 8-15
  const int l15  = lane & 15;

  // Preload all 32 A K-tiles (codebook rows, zero-padded past row 5).
  v2f a[32];
  const bool mvalid = (l15 < 6);
#pragma unroll
  for (int k = 0; k < 32; ++k) {
    const int kb = 4 * k + 2 * half;
    a[k].x = mvalid ? cb[l15 * DIN_ + kb + 0] : 0.0f;
    a[k].y = mvalid ? cb[l15 * DIN_ + kb + 1] : 0.0f;
  }

  float partial[6] = {0.f, 0.f, 0.f, 0.f, 0.f, 0.f};

  for (int j = 0; j < 16; ++j) {          // 16 tiles of 16 columns -> DH = 256
    const int nb = 16 * j;
    v8f c = {};
#pragma unroll
    for (int k = 0; k < 32; ++k) {        // 32 K-steps of 4 -> DIN = 128
      const int kb = 4 * k + 2 * half;
      v2f bv;
      bv.x = W1[(kb + 0) * DH_ + nb + l15];
      bv.y = W1[(kb + 1) * DH_ + nb + l15];
      c = __builtin_amdgcn_wmma_f32_16x16x4_f32(false, a[k], false, bv,
                                                (short)0, c, false, false);
    }
    // D layout: lanes 0-15 hold rows 0-7 (VGPR r = row r), col = nb + l15.
    const int   col  = nb + l15;
    const float bb   = b1[col];
    const float w2   = W2[col];
    const float gate = (half == 0) ? 1.0f : 0.0f;  // rows 0-5 live in lanes 0-15
#pragma unroll
    for (int mm = 0; mm < 6; ++mm) {
      float h = c[mm] + bb;
      h = h > 0.0f ? h : 0.0f;            // ReLU
      partial[mm] += gate * h * w2;
    }
  }

  // wave32 reduction of the W2 dot product; lanes 16-31 contribute zero.
#pragma unroll
  for (int mm = 0; mm < 6; ++mm) {
    float v = partial[mm];
    for (int off = 16; off > 0; off >>= 1) v += __shfl_xor(v, off, 32);
    if (lane == 0) s_out[mm] = v + b2[0];
  }
}

// ---------------------------------------------------------------------------
// Kernel 2: input_X = concat(tst_token, features), float4 (B128) path.
// 16400 rows x 32 float4 each; flat index over 524800 vectors.
// ---------------------------------------------------------------------------
__global__ __launch_bounds__(256)
void build_x(const float* __restrict__ feat, const float* __restrict__ tst,
             float* __restrict__ outX, int total4) {
  const int g = blockIdx.x * blockDim.x + threadIdx.x;
  if (g >= total4) return;
  const int row = g >> 5;                 // /32: b*1025 + i
  const int c4  = g & 31;
  const int b   = row / NP1;
  const int i   = row % NP1;
  const float4* src = (i == 0)
      ? ((const float4*)tst)
      : ((const float4*)(feat + ((size_t)b * N_ + (i - 1)) * DIN_));
  ((float4*)(outX + (size_t)row * DIN_))[c4] = src[c4];
}

// ---------------------------------------------------------------------------
// Kernel 3: input_MASK = concat(ones, mask) as float.
// ---------------------------------------------------------------------------
__global__ void build_mask(const uint8_t* __restrict__ mask,
                           float* __restrict__ outM, int total) {
  const int t = blockIdx.x * blockDim.x + threadIdx.x;
  if (t >= total) return;
  const int b = t / NP1, i = t % NP1;
  outM[t] = (i == 0) ? 1.0f : (mask[b * N_ + (i - 1)] ? 1.0f : 0.0f);
}

// ---------------------------------------------------------------------------
// Kernel 4: bias. One 256-thread block per output row (bi = b*1025 + i).
// The source row of Dm and the +1 edge adjustment are UNIFORM per block
// (scalar branch, no divergence). Each thread does one aligned int4
// (global_load_b128) covering j = 1+4t .. 4+4t; thread 0 also writes j = 0.
// Bucket map: D==k (k<5) -> s[k]; D==9999 -> s[5]; else 0.
// ---------------------------------------------------------------------------
__device__ __forceinline__ float bucket(int D, float s0, float s1, float s2,
                                        float s3, float s4, float s5) {
  return (D == 0) ? s0 : (D == 1) ? s1 : (D == 2) ? s2 :
         (D == 3) ? s3 : (D == 4) ? s4 : (D == 9999) ? s5 : 0.0f;
}

__global__ __launch_bounds__(256)
void build_bias(const int* __restrict__ Dm, const float* __restrict__ s,
                float* __restrict__ outB) {
  const int t  = threadIdx.x;             // 0..255
  const int bi = blockIdx.x;              // 0..16399
  const int b  = bi / NP1;
  const int i  = bi % NP1;

  const float s0 = s[0], s1 = s[1], s2 = s[2], s3 = s[3], s4 = s[4], s5 = s[5];

  // Uniform per block: which Dm row feeds this output row, and edge +1.
  const int  r    = (i == 0) ? 0 : (i - 1);
  const int  addv = (i == 0) ? 1 : 0;     // row 0 uses D[b,0,:]+1 (sentinel kept)
  const int* drow = Dm + ((size_t)b * N_ + r) * N_;

  float* orow = outB + (size_t)bi * NP1;

  // j = 0 (and the [0,0] corner which is D=0 -> s0).
  if (t == 0) {
    float v;
    if (i == 0) {
      v = s0;
    } else {
      const int d = Dm[(size_t)b * N_ * N_ + (i - 1)];   // D[b,0,i-1]
      const int D = (d == 9999) ? 9999 : d + 1;
      v = bucket(D, s0, s1, s2, s3, s4, s5);
    }
    orow[0] = v;
  }

  // j = 1+4t .. 4+4t ; source offset (j-1) = 4t -> 16B-aligned int4.
  const int4 d4 = *(const int4*)(drow + 4 * t);
  const int D0 = (d4.x == 9999) ? 9999 : d4.x + addv;
  const int D1 = (d4.y == 9999) ? 9999 : d4.y + addv;
  const int D2 = (d4.z == 9999) ? 9999 : d4.z + addv;
  const int D3 = (d4.w == 9999) ? 9999 : d4.w + addv;

  const int j0 = 1 + 4 * t;
  orow[j0 + 0] = bucket(D0, s0, s1, s2, s3, s4, s5);
  orow[j0 + 1] = bucket(D1, s0, s1, s2, s3, s4, s5);
  orow[j0 + 2] = bucket(D2, s0, s1, s2, s3, s4, s5);
  orow[j0 + 3] = bucket(D3, s0, s1, s2, s3, s4, s5);
}

// ---------------------------------------------------------------------------
extern "C" void kernel_launch(void* const* d_in, const int* in_sizes, int n_in,
                              void* d_out, int out_size, void* d_ws, size_t ws_size,
                              hipStream_t stream) {
  const float*   feat = (const float*)d_in[0];
  const uint8_t* mask = (const uint8_t*)d_in[1];
  const int*     Dm   = (const int*)d_in[2];
  const float*   tst  = (const float*)d_in[3];
  const float*   cb   = (const float*)d_in[4];
  const float*   W1   = (const float*)d_in[5];
  const float*   b1   = (const float*)d_in[6];
  const float*   W2   = (const float*)d_in[7];
  const float*   b2   = (const float*)d_in[8];

  float* outX = (float*)d_out;                       // 16*1025*128
  float* outM = outX + (size_t)B_ * NP1 * DIN_;      // 16*1025
  float* outB = outM + (size_t)B_ * NP1;             // 16*1025*1025
  float* s    = (float*)d_ws;                        // 6 bucket scalars

  mlp6_wmma<<<1, 32, 0, stream>>>(cb, W1, b1, W2, b2, s);

  const int total4 = B_ * NP1 * (DIN_ / 4);          // 524800 float4s
  build_x<<<(total4 + 255) / 256, 256, 0, stream>>>(feat, tst, outX, total4);

  const int mtot = B_ * NP1;
  build_mask<<<(mtot + 255) / 256, 256, 0, stream>>>(mask, outM, mtot);

  build_bias<<<B_ * NP1, 256, 0, stream>>>(Dm, s, outB);
}